// LesionTokenBuilder_9560597201600
// MI455X (gfx1250) — compile-verified
//
#include <hip/hip_runtime.h>
#include <math.h>

#define BATCH 256
#define NQ    900
#define DIM   256
#define NCLS  15      // C1-1 classes participate in max
#define TOPK  20
#define LN_EPS 1e-5f

#define NTOK   (BATCH * TOPK)          // 5120 tokens
#define OUT_TOKENS (NTOK * DIM)        // 1310720 floats
#define H1STRIDE 516                   // 512 cols padded (4*516 % 64banks -> conflict free)
#define TOKSTRIDE 260                  // 256 cols padded

typedef __attribute__((ext_vector_type(2))) float v2f;
typedef __attribute__((ext_vector_type(8))) float v8f;

__device__ __forceinline__ float gelu_erf(float x) {
    return 0.5f * x * (1.0f + erff(x * 0.70710678118654752440f));
}

// ---------------------------------------------------------------------------
// Kernel A: per-batch scores + top-20 selection (ties -> lowest index)
// ---------------------------------------------------------------------------
__global__ void __launch_bounds__(256)
topk_kernel(const float* __restrict__ logits,   // [B, Q, 16]
            float* __restrict__ out_f,          // d_out as float
            int*   __restrict__ out_i,          // d_out as int
            int*   __restrict__ ws_idx,         // [B*K]
            float* __restrict__ ws_sc)          // [B*K]
{
    __shared__ float sc[NQ];
    __shared__ float rv[256];
    __shared__ int   ri[256];

    const int b   = blockIdx.x;
    const int tid = threadIdx.x;

    // scores = sigmoid(max over 15 classes)  (sigmoid monotonic)
    for (int q = tid; q < NQ; q += 256) {
        const float* lg = logits + ((size_t)b * NQ + q) * 16;
        float m = lg[0];
        #pragma unroll
        for (int c = 1; c < NCLS; ++c) m = fmaxf(m, lg[c]);
        sc[q] = 1.0f / (1.0f + expf(-m));
    }
    __syncthreads();

    for (int it = 0; it < TOPK; ++it) {
        // local argmax over strided subset (keeps smallest index on ties)
        float bv = -INFINITY; int bi = 0;
        for (int q = tid; q < NQ; q += 256) {
            float v = sc[q];
            if (v > bv) { bv = v; bi = q; }
        }
        rv[tid] = bv; ri[tid] = bi;
        __syncthreads();
        // tree reduce: larger value wins; equal value -> smaller index wins
        for (int off = 128; off > 0; off >>= 1) {
            if (tid < off) {
                float ov = rv[tid + off]; int oi = ri[tid + off];
                if (ov > rv[tid] || (ov == rv[tid] && oi < ri[tid])) {
                    rv[tid] = ov; ri[tid] = oi;
                }
            }
            __syncthreads();
        }
        if (tid == 0) {
            int   wi = ri[0];
            float wv = rv[0];
            out_f[OUT_TOKENS + b * TOPK + it]        = wv;      // topk_scores
            out_i[OUT_TOKENS + NTOK + b * TOPK + it] = wi;      // topk_idx (raw i32)
            ws_sc[b * TOPK + it]  = wv;
            ws_idx[b * TOPK + it] = wi;
            sc[wi] = -INFINITY;
        }
        __syncthreads();
    }
}

// ---------------------------------------------------------------------------
// Kernel B: gather + fused MLPs (WMMA f32 16x16x4, K=512) + LayerNorm
// Block = 512 threads (16 waves); 16 tokens per block; wave w owns N-tile w.
// ---------------------------------------------------------------------------
__global__ void __launch_bounds__(512)
tokens_kernel(const float* __restrict__ hidden,    // [B, Q, D]
              const float* __restrict__ boxes,     // [B, Q, 4]
              const float* __restrict__ box_w1,    // [4, D]
              const float* __restrict__ box_b1,    // [D]
              const float* __restrict__ box_w2,    // [D, D]
              const float* __restrict__ box_b2,    // [D]
              const float* __restrict__ score_w1,  // [1, D]
              const float* __restrict__ score_b1,  // [D]
              const float* __restrict__ score_w2,  // [D, D]
              const float* __restrict__ score_b2,  // [D]
              const float* __restrict__ ln_g,      // [D]
              const float* __restrict__ ln_b,      // [D]
              const int*   __restrict__ ws_idx,    // [B*K]
              const float* __restrict__ ws_sc,     // [B*K]
              float* __restrict__ out)             // [B*K, D]
{
    __shared__ float h1[16 * H1STRIDE];    // gelu hidden, 16 tokens x 512 (box|score)
    __shared__ float tok[16 * TOKSTRIDE];  // pre-LN tokens, 16 x 256
    __shared__ int   s_idx[16];
    __shared__ float s_sc[16];
    __shared__ float s_box[16][4];

    const int tid = threadIdx.x;
    const int t0  = blockIdx.x * 16;

    // preload per-token metadata
    if (tid < 16) {
        int t = t0 + tid;
        int b = t / TOPK, j = t % TOPK;
        int idx = ws_idx[b * TOPK + j];
        s_idx[tid] = idx;
        s_sc[tid]  = ws_sc[b * TOPK + j];
        const float* bx = boxes + ((size_t)b * NQ + idx) * 4;
        #pragma unroll
        for (int i = 0; i < 4; ++i) s_box[tid][i] = bx[i];
    }
    __syncthreads();

    // ---- Phase 1: first MLP layers + exact-erf GELU into LDS -------------
    {
        const int k = tid;  // 0..511
        for (int m = 0; m < 16; ++m) {
            float x;
            if (k < DIM) {
                x = box_b1[k];
                #pragma unroll
                for (int i = 0; i < 4; ++i) x += s_box[m][i] * box_w1[i * DIM + k];
            } else {
                int kk = k - DIM;
                x = s_sc[m] * score_w1[kk] + score_b1[kk];
            }
            h1[m * H1STRIDE + k] = gelu_erf(x);
        }
    }
    __syncthreads();

    // ---- Phase 2: WMMA f32 16x16x4, K = 512 (box 256 | score 256) --------
    {
        const int wave = tid >> 5;          // 0..15 -> N-tile
        const int lane = tid & 31;
        const int half = lane >> 4;         // 0 | 1
        const int nl   = lane & 15;
        const int n    = wave * 16 + nl;    // global output column
        const int mrow = nl;                // A-matrix row index for this lane

        v8f acc = {};

        // box_w2 part: K = 0..255 of h1
        #pragma unroll 4
        for (int kb = 0; kb < DIM; kb += 4) {
            int kk = kb + 2 * half;
            v2f a, bb;
            a.x  = h1[mrow * H1STRIDE + kk];
            a.y  = h1[mrow * H1STRIDE + kk + 1];
            bb.x = box_w2[(size_t)kk * DIM + n];
            bb.y = box_w2[(size_t)(kk + 1) * DIM + n];
            acc = __builtin_amdgcn_wmma_f32_16x16x4_f32(
                false, a, false, bb, (short)0, acc, false, false);
        }
        // score_w2 part: K = 256..511 of h1
        #pragma unroll 4
        for (int kb = 0; kb < DIM; kb += 4) {
            int kk = kb + 2 * half;
            v2f a, bb;
            a.x  = h1[mrow * H1STRIDE + DIM + kk];
            a.y  = h1[mrow * H1STRIDE + DIM + kk + 1];
            bb.x = score_w2[(size_t)kk * DIM + n];
            bb.y = score_w2[(size_t)(kk + 1) * DIM + n];
            acc = __builtin_amdgcn_wmma_f32_16x16x4_f32(
                false, a, false, bb, (short)0, acc, false, false);
        }

        // C/D layout: vgpr i, lanes 0-15 -> M=i, lanes 16-31 -> M=i+8; N=lane%16
        const float b2 = box_b2[n] + score_b2[n];
        #pragma unroll
        for (int i = 0; i < 8; ++i) {
            int row = i + 8 * half;
            tok[row * TOKSTRIDE + n] = acc[i] + b2;
        }
    }
    __syncthreads();

    // ---- Phase 3: add gathered hidden + LayerNorm (one wave per row) -----
    {
        const int w    = tid >> 5;   // row 0..15
        const int lane = tid & 31;
        const int t    = t0 + w;
        const int b    = t / TOPK;
        const int idx  = s_idx[w];
        const float* hid = hidden + ((size_t)b * NQ + idx) * DIM;

        float v[8];
        float s = 0.0f, s2 = 0.0f;
        #pragma unroll
        for (int j = 0; j < 8; ++j) {
            int col = lane + 32 * j;
            float val = tok[w * TOKSTRIDE + col] + hid[col];
            v[j] = val;
            s  += val;
            s2 += val * val;
        }
        #pragma unroll
        for (int off = 16; off > 0; off >>= 1) {
            s  += __shfl_xor(s,  off, 32);
            s2 += __shfl_xor(s2, off, 32);
        }
        const float mu  = s * (1.0f / DIM);
        const float var = s2 * (1.0f / DIM) - mu * mu;
        const float inv = rsqrtf(var + LN_EPS);

        float* op = out + (size_t)t * DIM;
        #pragma unroll
        for (int j = 0; j < 8; ++j) {
            int col = lane + 32 * j;
            op[col] = (v[j] - mu) * inv * ln_g[col] + ln_b[col];
        }
    }
}

// ---------------------------------------------------------------------------
extern "C" void kernel_launch(void* const* d_in, const int* in_sizes, int n_in,
                              void* d_out, int out_size, void* d_ws, size_t ws_size,
                              hipStream_t stream) {
    const float* lesion_hidden = (const float*)d_in[0];
    const float* lesion_boxes  = (const float*)d_in[1];
    const float* lesion_logits = (const float*)d_in[2];
    const float* box_w1   = (const float*)d_in[3];
    const float* box_b1   = (const float*)d_in[4];
    const float* box_w2   = (const float*)d_in[5];
    const float* box_b2   = (const float*)d_in[6];
    const float* score_w1 = (const float*)d_in[7];
    const float* score_b1 = (const float*)d_in[8];
    const float* score_w2 = (const float*)d_in[9];
    const float* score_b2 = (const float*)d_in[10];
    const float* ln_g     = (const float*)d_in[11];
    const float* ln_b     = (const float*)d_in[12];

    float* out_f  = (float*)d_out;
    int*   out_i  = (int*)d_out;
    int*   ws_idx = (int*)d_ws;
    float* ws_sc  = (float*)((char*)d_ws + NTOK * sizeof(int));

    topk_kernel<<<BATCH, 256, 0, stream>>>(lesion_logits, out_f, out_i, ws_idx, ws_sc);

    tokens_kernel<<<NTOK / 16, 512, 0, stream>>>(
        lesion_hidden, lesion_boxes,
        box_w1, box_b1, box_w2, box_b2,
        score_w1, score_b1, score_w2, score_b2,
        ln_g, ln_b, ws_idx, ws_sc, out_f);
}